// AttentionalGNN_44633300140734
// MI455X (gfx1250) — compile-verified
//
#include <hip/hip_runtime.h>

// ---------------------------------------------------------------------------
// AttentionalGNN forward for MI455X (gfx1250), wave32, WMMA f32 16x16x4.
// Memory-bound problem (dist = 75.5MB f32, ~10 GFLOP total) -> keep f32
// precision (free on the roofline) and fuse the cross-attn score epilogue
// into the big W1d x dist GEMM so the 75.5MB hidden tensor is never stored.
// Round 2: branch-free, software-pipelined WMMA inner loops.
// ---------------------------------------------------------------------------

#define NDESC 384
#define NHALF 192
#define NPAIR (384 * 384)

typedef __attribute__((ext_vector_type(2))) float v2f;
typedef __attribute__((ext_vector_type(8))) float v8f;

__device__ __forceinline__ v8f wmma4(v2f a, v2f b, v8f c) {
  // 8-arg pattern: (neg_a, A, neg_b, B, c_mod, C, reuse_a, reuse_b)
  return __builtin_amdgcn_wmma_f32_16x16x4_f32(false, a, false, b, (short)0, c,
                                               false, false);
}

// ---------------------------------------------------------------------------
// Generic f32 GEMM: C[M,N] = A[M,K] x B (+bias[M]) (relu?) (+addC)
// A row-major stride lda. B: row-major [K,N] stride ldb, or (TB) row-major
// [N,K] stride ldb (prob^T in the message GEMMs). Block = 256 threads =
// 8 waves; wave w computes rows [by*128+16w, +16) x 64 columns via 4
// accumulators, K stepped by 4 (f32 WMMA). Software-pipelined: next K-step's
// fragments are loaded before the current WMMAs issue.
// Requires: M % 128 == 0, N % 64 == 0, K % 4 == 0, K >= 8.
// ---------------------------------------------------------------------------
template <bool TB>
__global__ __launch_bounds__(256) void k_gemm(
    const float* __restrict__ A, int lda, const float* __restrict__ B, int ldb,
    const float* __restrict__ bias, const float* __restrict__ addC,
    float* __restrict__ C, int ldc, int K, int relu) {
  const int lane = threadIdx.x & 31;
  const int wv = threadIdx.x >> 5;
  const int lr = lane & 15;
  const int hi = lane >> 4;
  const int m0 = blockIdx.y * 128 + wv * 16;
  const int n0 = blockIdx.x * 64;

  v8f acc[4];
#pragma unroll
  for (int t = 0; t < 4; ++t)
    acc[t] = (v8f){0.f, 0.f, 0.f, 0.f, 0.f, 0.f, 0.f, 0.f};

  // A fragment per ISA 16x4 layout: lanes 0-15 hold K=k..k+1, lanes 16-31
  // hold K=k+2..k+3 of rows m0+lr.
  const float* arow = A + (size_t)(m0 + lr) * lda + 2 * hi;

  auto loadB = [&](int k, int t) {
    v2f b;
    if (TB) {
      const float* bp = B + (size_t)(n0 + 16 * t + lr) * ldb + k + 2 * hi;
      b.x = bp[0];
      b.y = bp[1];
    } else {
      const float* bp = B + (size_t)(k + 2 * hi) * ldb + n0 + 16 * t + lr;
      b.x = bp[0];
      b.y = bp[ldb];
    }
    return b;
  };

  // prologue: fragments for k = 0
  v2f a_c, b_c[4];
  a_c.x = arow[0];
  a_c.y = arow[1];
#pragma unroll
  for (int t = 0; t < 4; ++t) b_c[t] = loadB(0, t);

#pragma unroll 2
  for (int k = 0; k < K - 4; k += 4) {
    v2f a_n, b_n[4];
    a_n.x = arow[k + 4];
    a_n.y = arow[k + 5];
#pragma unroll
    for (int t = 0; t < 4; ++t) b_n[t] = loadB(k + 4, t);
#pragma unroll
    for (int t = 0; t < 4; ++t) acc[t] = wmma4(a_c, b_c[t], acc[t]);
    a_c = a_n;
#pragma unroll
    for (int t = 0; t < 4; ++t) b_c[t] = b_n[t];
  }
#pragma unroll
  for (int t = 0; t < 4; ++t) acc[t] = wmma4(a_c, b_c[t], acc[t]);

  // C layout: VGPR e -> row m0 + 8*hi + e, col n0+16t+lr
#pragma unroll
  for (int t = 0; t < 4; ++t) {
    const int col = n0 + 16 * t + lr;
#pragma unroll
    for (int e = 0; e < 8; ++e) {
      const int row = m0 + 8 * hi + e;
      float vl = acc[t][e];
      if (bias) vl += bias[row];
      if (relu) vl = fmaxf(vl, 0.f);
      if (addC) vl += addC[(size_t)row * ldc + col];
      C[(size_t)row * ldc + col] = vl;
    }
  }
}

// ---------------------------------------------------------------------------
// Fused cross-attention scores. One block = 64 consecutive dist positions
// p = r*384 + c (r fixed per block). Computes Hd = W1d x dist tile (128x64)
// with WMMA (K = 128, fully unrolled + pipelined), then for both directions
//   score1[r,c] = b2 + sum_h W2[h]*relu(Hd[h]+A1[h,r]+B1[h,c]+b1[h])
//   score0[c,r] = b2 + sum_h W2[h]*relu(Hd[h]+A0[h,c]+B0[h,r]+b1[h])
// reduced across the 8 waves through LDS. Hd never touches memory.
// ---------------------------------------------------------------------------
__global__ __launch_bounds__(256) void k_cross_score(
    const float* __restrict__ W1, const float* __restrict__ b1,
    const float* __restrict__ W2, const float* __restrict__ b2,
    const float* __restrict__ dist, const float* __restrict__ A1,
    const float* __restrict__ B1, const float* __restrict__ A0,
    const float* __restrict__ B0, float* __restrict__ s1,
    float* __restrict__ s0) {
  __shared__ float red1[64];
  __shared__ float red0[64];
  const int lane = threadIdx.x & 31;
  const int wv = threadIdx.x >> 5;
  const int lr = lane & 15;
  const int hi = lane >> 4;
  const int p0 = blockIdx.x * 64;
  const int r = p0 / NDESC;
  const int c0 = p0 % NDESC;

  if (threadIdx.x < 64) {
    red1[threadIdx.x] = 0.f;
    red0[threadIdx.x] = 0.f;
  }
  __syncthreads();

  v8f acc[4];
#pragma unroll
  for (int t = 0; t < 4; ++t)
    acc[t] = (v8f){0.f, 0.f, 0.f, 0.f, 0.f, 0.f, 0.f, 0.f};

  // W1d = columns 256..383 of W1[128x384]
  const float* arow = W1 + (size_t)(wv * 16 + lr) * 384 + 256 + 2 * hi;
  const float* brow = dist + (size_t)(2 * hi) * NPAIR + p0 + lr;

  auto loadB = [&](int k, int t) {
    v2f b;
    const float* bp = brow + (size_t)k * NPAIR + 16 * t;
    b.x = bp[0];
    b.y = bp[NPAIR];
    return b;
  };

  v2f a_c, b_c[4];
  a_c.x = arow[0];
  a_c.y = arow[1];
#pragma unroll
  for (int t = 0; t < 4; ++t) b_c[t] = loadB(0, t);

#pragma unroll
  for (int k = 0; k < 124; k += 4) {
    v2f a_n, b_n[4];
    a_n.x = arow[k + 4];
    a_n.y = arow[k + 5];
#pragma unroll
    for (int t = 0; t < 4; ++t) b_n[t] = loadB(k + 4, t);
#pragma unroll
    for (int t = 0; t < 4; ++t) acc[t] = wmma4(a_c, b_c[t], acc[t]);
    a_c = a_n;
#pragma unroll
    for (int t = 0; t < 4; ++t) b_c[t] = b_n[t];
  }
#pragma unroll
  for (int t = 0; t < 4; ++t) acc[t] = wmma4(a_c, b_c[t], acc[t]);

  const int hb = wv * 16 + 8 * hi;  // first of this lane's 8 hidden channels
  float w2v[8], b1v[8], a1v[8], b0v[8];
#pragma unroll
  for (int e = 0; e < 8; ++e) {
    const int h = hb + e;
    w2v[e] = W2[h];
    b1v[e] = b1[h];
    a1v[e] = A1[(size_t)h * NDESC + r];
    b0v[e] = B0[(size_t)h * NDESC + r];
  }
#pragma unroll
  for (int t = 0; t < 4; ++t) {
    const int c = c0 + 16 * t + lr;
    float p1 = 0.f, q0 = 0.f;
#pragma unroll
    for (int e = 0; e < 8; ++e) {
      const int h = hb + e;
      const float base = acc[t][e] + b1v[e];
      const float x1 = base + a1v[e] + B1[(size_t)h * NDESC + c];
      const float x0 = base + b0v[e] + A0[(size_t)h * NDESC + c];
      p1 += w2v[e] * fmaxf(x1, 0.f);
      q0 += w2v[e] * fmaxf(x0, 0.f);
    }
    atomicAdd(&red1[16 * t + lr], p1);
    atomicAdd(&red0[16 * t + lr], q0);
  }
  __syncthreads();
  if (threadIdx.x < 64) {
    const int c = c0 + threadIdx.x;
    const float bb = b2[0];
    s1[(size_t)r * NDESC + c] = red1[threadIdx.x] + bb;
    s0[(size_t)c * NDESC + r] = red0[threadIdx.x] + bb;
  }
}

// ---------------------------------------------------------------------------
// Multi-head attention core (tiny FLOPs, VALU). q/k/v are projected buffers
// [128 x Nq/Nk]. Channel d of head h lives at row d*4+h (jax reshape order).
// One block per (query n, head h). Writes merged x[128 x Nq].
// ---------------------------------------------------------------------------
__global__ __launch_bounds__(128) void k_mha(const float* __restrict__ q,
                                             const float* __restrict__ k,
                                             const float* __restrict__ v,
                                             float* __restrict__ x, int Nq,
                                             int Nk) {
  __shared__ float qh[32];
  __shared__ float sc[384];
  __shared__ float red[128];
  const int h = blockIdx.y;
  const int n = blockIdx.x;
  const int tid = threadIdx.x;
  if (tid < 32) qh[tid] = q[(size_t)(tid * 4 + h) * Nq + n];
  __syncthreads();
  const float scale = 0.17677669529663687f;  // 1/sqrt(32)
  float lmax = -3.0e38f;
  for (int m = tid; m < Nk; m += 128) {
    float s = 0.f;
    for (int d = 0; d < 32; ++d) s += qh[d] * k[(size_t)(d * 4 + h) * Nk + m];
    s *= scale;
    sc[m] = s;
    lmax = fmaxf(lmax, s);
  }
  red[tid] = lmax;
  __syncthreads();
  for (int off = 64; off > 0; off >>= 1) {
    if (tid < off) red[tid] = fmaxf(red[tid], red[tid + off]);
    __syncthreads();
  }
  const float mx = red[0];
  __syncthreads();
  float lsum = 0.f;
  for (int m = tid; m < Nk; m += 128) {
    const float e = expf(sc[m] - mx);
    sc[m] = e;
    lsum += e;
  }
  red[tid] = lsum;
  __syncthreads();
  for (int off = 64; off > 0; off >>= 1) {
    if (tid < off) red[tid] += red[tid + off];
    __syncthreads();
  }
  const float inv = 1.f / red[0];
  __syncthreads();
  const int hd = tid >> 2, part = tid & 3;
  const int seg = Nk >> 2;
  float a = 0.f;
  const float* vp = v + (size_t)(hd * 4 + h) * Nk;
  for (int m = part * seg; m < part * seg + seg; ++m) a += sc[m] * vp[m];
  red[tid] = a;
  __syncthreads();
  if (part == 0)
    x[(size_t)(hd * 4 + h) * Nq + n] =
        (red[tid] + red[tid + 1] + red[tid + 2] + red[tid + 3]) * inv;
}

// Row-wise softmax (prob over keys), one block per row.
__global__ __launch_bounds__(256) void k_row_softmax(const float* __restrict__ s,
                                                     float* __restrict__ p,
                                                     int N) {
  __shared__ float red[256];
  const int r = blockIdx.x, tid = threadIdx.x;
  const float* sr = s + (size_t)r * N;
  float* pr = p + (size_t)r * N;
  float lmax = -3.0e38f;
  for (int c = tid; c < N; c += 256) lmax = fmaxf(lmax, sr[c]);
  red[tid] = lmax;
  __syncthreads();
  for (int off = 128; off > 0; off >>= 1) {
    if (tid < off) red[tid] = fmaxf(red[tid], red[tid + off]);
    __syncthreads();
  }
  const float mx = red[0];
  __syncthreads();
  float ls = 0.f;
  for (int c = tid; c < N; c += 256) {
    const float e = expf(sr[c] - mx);
    pr[c] = e;
    ls += e;
  }
  red[tid] = ls;
  __syncthreads();
  for (int off = 128; off > 0; off >>= 1) {
    if (tid < off) red[tid] += red[tid + off];
    __syncthreads();
  }
  const float inv = 1.f / red[0];
  __syncthreads();
  for (int c = tid; c < N; c += 256) pr[c] *= inv;
}

// Column-wise log_softmax (axis=-2 of score1), one block per column.
__global__ __launch_bounds__(256) void k_col_logsoftmax(
    const float* __restrict__ s, float* __restrict__ o, int Nr, int Nc) {
  __shared__ float red[256];
  const int col = blockIdx.x, tid = threadIdx.x;
  float lmax = -3.0e38f;
  for (int r = tid; r < Nr; r += 256)
    lmax = fmaxf(lmax, s[(size_t)r * Nc + col]);
  red[tid] = lmax;
  __syncthreads();
  for (int off = 128; off > 0; off >>= 1) {
    if (tid < off) red[tid] = fmaxf(red[tid], red[tid + off]);
    __syncthreads();
  }
  const float mx = red[0];
  __syncthreads();
  float ls = 0.f;
  for (int r = tid; r < Nr; r += 256) ls += expf(s[(size_t)r * Nc + col] - mx);
  red[tid] = ls;
  __syncthreads();
  for (int off = 128; off > 0; off >>= 1) {
    if (tid < off) red[tid] += red[tid + off];
    __syncthreads();
  }
  const float lse = mx + logf(red[0]);
  __syncthreads();
  for (int r = tid; r < Nr; r += 256)
    o[(size_t)r * Nc + col] = s[(size_t)r * Nc + col] - lse;
}

__global__ void k_copy(const float* __restrict__ s, float* __restrict__ d,
                       int n) {
  const int i = blockIdx.x * 256 + threadIdx.x;
  if (i < n) d[i] = s[i];
}
__global__ void k_zero(float* __restrict__ p, int n) {
  const int i = blockIdx.x * 256 + threadIdx.x;
  if (i < n) p[i] = 0.f;
}

__global__ void k_couplings(const float* __restrict__ s,
                            const float* __restrict__ alpha,
                            float* __restrict__ coup) {
  const int i = blockIdx.x * 256 + threadIdx.x;
  if (i >= 385 * 385) return;
  const int r = i / 385, c = i % 385;
  coup[i] = (r < 384 && c < 384) ? s[(size_t)r * 384 + c] : alpha[0];
}

__global__ __launch_bounds__(256) void k_sink_row(
    const float* __restrict__ coup, const float* __restrict__ v,
    float* __restrict__ u) {
  __shared__ float red[256];
  const int r = blockIdx.x, tid = threadIdx.x;
  const float* cr = coup + (size_t)r * 385;
  float lmax = -3.0e38f;
  for (int c = tid; c < 385; c += 256) lmax = fmaxf(lmax, cr[c] + v[c]);
  red[tid] = lmax;
  __syncthreads();
  for (int off = 128; off > 0; off >>= 1) {
    if (tid < off) red[tid] = fmaxf(red[tid], red[tid + off]);
    __syncthreads();
  }
  const float mx = red[0];
  __syncthreads();
  float ls = 0.f;
  for (int c = tid; c < 385; c += 256) ls += expf(cr[c] + v[c] - mx);
  red[tid] = ls;
  __syncthreads();
  for (int off = 128; off > 0; off >>= 1) {
    if (tid < off) red[tid] += red[tid + off];
    __syncthreads();
  }
  if (tid == 0) {
    const float norm = -logf(768.f);
    const float lmu = (r < 384) ? norm : (logf(384.f) + norm);
    u[r] = lmu - (mx + logf(red[0]));
  }
}

__global__ __launch_bounds__(256) void k_sink_col(
    const float* __restrict__ coup, const float* __restrict__ u,
    float* __restrict__ v) {
  __shared__ float red[256];
  const int c = blockIdx.x, tid = threadIdx.x;
  float lmax = -3.0e38f;
  for (int r = tid; r < 385; r += 256)
    lmax = fmaxf(lmax, coup[(size_t)r * 385 + c] + u[r]);
  red[tid] = lmax;
  __syncthreads();
  for (int off = 128; off > 0; off >>= 1) {
    if (tid < off) red[tid] = fmaxf(red[tid], red[tid + off]);
    __syncthreads();
  }
  const float mx = red[0];
  __syncthreads();
  float ls = 0.f;
  for (int r = tid; r < 385; r += 256)
    ls += expf(coup[(size_t)r * 385 + c] + u[r] - mx);
  red[tid] = ls;
  __syncthreads();
  for (int off = 128; off > 0; off >>= 1) {
    if (tid < off) red[tid] += red[tid + off];
    __syncthreads();
  }
  if (tid == 0) {
    const float norm = -logf(768.f);
    const float lnu = (c < 384) ? norm : (logf(384.f) + norm);
    v[c] = lnu - (mx + logf(red[0]));
  }
}

__global__ __launch_bounds__(256) void k_zmake(
    const float* __restrict__ coup, const float* __restrict__ u,
    const float* __restrict__ v, float* __restrict__ z,
    float* __restrict__ pmin, float* __restrict__ pmax) {
  __shared__ float rmn[256], rmx[256];
  const int tid = threadIdx.x;
  const int i = blockIdx.x * 256 + tid;  // 576*256 == 147456 exactly
  const int r = i / 384, c = i % 384;
  const float norm = -logf(768.f);
  const float zz = coup[(size_t)r * 385 + c] + u[r] + v[c] - norm;
  z[i] = zz;
  rmn[tid] = zz;
  rmx[tid] = zz;
  __syncthreads();
  for (int off = 128; off > 0; off >>= 1) {
    if (tid < off) {
      rmn[tid] = fminf(rmn[tid], rmn[tid + off]);
      rmx[tid] = fmaxf(rmx[tid], rmx[tid + off]);
    }
    __syncthreads();
  }
  if (tid == 0) {
    pmin[blockIdx.x] = rmn[0];
    pmax[blockIdx.x] = rmx[0];
  }
}

__global__ __launch_bounds__(256) void k_minmax(const float* __restrict__ pmin,
                                                const float* __restrict__ pmax,
                                                int n, float* __restrict__ smin) {
  __shared__ float rmn[256], rmx[256];
  const int tid = threadIdx.x;
  float a = 3.0e38f, b = -3.0e38f;
  for (int i = tid; i < n; i += 256) {
    a = fminf(a, pmin[i]);
    b = fmaxf(b, pmax[i]);
  }
  rmn[tid] = a;
  rmx[tid] = b;
  __syncthreads();
  for (int off = 128; off > 0; off >>= 1) {
    if (tid < off) {
      rmn[tid] = fminf(rmn[tid], rmn[tid + off]);
      rmx[tid] = fmaxf(rmx[tid], rmx[tid + off]);
    }
    __syncthreads();
  }
  if (tid == 0) smin[0] = rmn[0] - rmx[0];
}

__global__ __launch_bounds__(256) void k_final(
    const float* __restrict__ z, const float* __restrict__ sminp,
    const int* __restrict__ lmb, const int* __restrict__ fidx,
    const unsigned char* __restrict__ unr, float* __restrict__ out) {
  const int i = blockIdx.x * 256 + threadIdx.x;
  if (i >= NPAIR) return;
  const int b = i / 384, f = i % 384;
  const float smin = sminp[0];
  const int f0 = fidx[2 * f], f1 = fidx[2 * f + 1];
  const bool inv = (f0 < lmb[4 * b]) || (f0 >= lmb[4 * b + 1]) ||
                   (f1 < lmb[4 * b + 2]) || (f1 >= lmb[4 * b + 3]);
  float val = z[i];
  if (inv) val += smin - 40.f;
  if (unr[i]) val += smin - 20.f;
  out[i] = val * 15.f;
}

// ---------------------------------------------------------------------------
// Host side: parameter offsets follow jax pytree flatten order
// (dict keys sorted alphabetically; lists keep order).
// ---------------------------------------------------------------------------
struct MlpP { size_t W1, W2, b1, b2; };
struct MhaP { size_t mW, mb, qW, qb, kW, kb, vW, vb; };
struct CrossP { size_t W1, W2, b1, b2; };

static size_t s_take(size_t& o, size_t n) { size_t r = o; o += n; return r; }
static MlpP s_mlp(size_t& o, size_t cin, size_t chid, size_t cout) {
  MlpP p;
  p.W1 = s_take(o, chid * cin);
  p.W2 = s_take(o, cout * chid);
  p.b1 = s_take(o, chid);
  p.b2 = s_take(o, cout);
  return p;
}
static MhaP s_mha(size_t& o) {
  MhaP p;
  p.mW = s_take(o, 16384); p.mb = s_take(o, 128);
  p.qW = s_take(o, 16384); p.qb = s_take(o, 128);
  p.kW = s_take(o, 16384); p.kb = s_take(o, 128);
  p.vW = s_take(o, 16384); p.vb = s_take(o, 128);
  return p;
}

extern "C" void kernel_launch(void* const* d_in, const int* in_sizes, int n_in,
                              void* d_out, int out_size, void* d_ws,
                              size_t ws_size, hipStream_t stream) {
  (void)in_sizes; (void)n_in; (void)out_size; (void)ws_size;
  const float* in_d0 = (const float*)d_in[0];
  const float* in_d1 = (const float*)d_in[1];
  const float* in_d2 = (const float*)d_in[2];
  const float* in_d3 = (const float*)d_in[3];
  const int* lmb = (const int*)d_in[4];
  const int* fidx = (const int*)d_in[5];
  const float* dist = (const float*)d_in[8];
  const unsigned char* unreach = (const unsigned char*)d_in[9];
  const float* P = (const float*)d_in[10];
  float* out = (float*)d_out;

  // ---- workspace layout (floats) ----
  float* ws = (float*)d_ws;
  size_t o = 0;
  auto alloc = [&](size_t n) { float* p = ws + o; o += n; return p; };
  float* d0a = alloc(49152); float* d0b = alloc(49152);
  float* d1a = alloc(49152); float* d1b = alloc(49152);
  float* d2a = alloc(24576); float* d2b = alloc(24576);
  float* d3a = alloc(24576); float* d3b = alloc(24576);
  float* qbuf = alloc(49152); float* kbuf = alloc(49152);
  float* vbuf = alloc(49152); float* xbuf = alloc(49152);
  float* concat = alloc(98304); float* hid = alloc(98304);
  float* A1b = alloc(49152); float* B1b = alloc(49152);
  float* A0b = alloc(49152); float* B0b = alloc(49152);
  float* s1b = alloc((size_t)NPAIR); float* s0b = alloc((size_t)NPAIR);
  float* prob = alloc((size_t)NPAIR);
  float* coup = alloc(385 * 385);
  float* uvec = alloc(385); float* vvec = alloc(385);
  float* sb = alloc((size_t)NPAIR); float* zb = alloc((size_t)NPAIR);
  float* pmin = alloc(576); float* pmax = alloc(576);
  float* sminb = alloc(1);

  // ---- parameter offsets ----
  size_t po = 0;
  const size_t bin_off = s_take(po, 1);
  const int is_cross[4] = {0, 1, 0, 1};
  CrossP ca[4]; MhaP sa[4]; MlpP am[4];
  MhaP gha[4]; MlpP ghm[4]; MhaP pha[4]; MlpP phm[4];
  for (int li = 0; li < 4; ++li) {
    if (is_cross[li]) {
      MlpP t = s_mlp(po, 384, 128, 1);
      ca[li].W1 = t.W1; ca[li].W2 = t.W2; ca[li].b1 = t.b1; ca[li].b2 = t.b2;
    } else {
      sa[li] = s_mha(po);
    }
    am[li] = s_mlp(po, 256, 256, 128);
    gha[li] = s_mha(po); ghm[li] = s_mlp(po, 256, 256, 128);
    pha[li] = s_mha(po); phm[li] = s_mlp(po, 256, 256, 128);
  }

  auto gemm = [&](const float* A, int lda, const float* B, int ldb, int tB,
                  const float* bias, const float* addC, float* C, int ldc,
                  int M, int N, int K, int relu) {
    dim3 g(N / 64, M / 128);
    if (tB)
      k_gemm<true><<<g, 256, 0, stream>>>(A, lda, B, ldb, bias, addC, C, ldc,
                                          K, relu);
    else
      k_gemm<false><<<g, 256, 0, stream>>>(A, lda, B, ldb, bias, addC, C, ldc,
                                           K, relu);
  };
  auto copyk = [&](const float* s, float* d, int n) {
    k_copy<<<(n + 255) / 256, 256, 0, stream>>>(s, d, n);
  };
  // self-style attention propagation: delta = mlp2([qdesc, mha]) ; out = addb + delta
  auto mha_prop = [&](const MhaP& mp, const MlpP& lp, const float* qd, int Nq,
                      const float* sd, int Nk, const float* addb, float* outd) {
    gemm(P + mp.qW, 128, qd, Nq, 0, P + mp.qb, nullptr, qbuf, Nq, 128, Nq, 128, 0);
    gemm(P + mp.kW, 128, sd, Nk, 0, P + mp.kb, nullptr, kbuf, Nk, 128, Nk, 128, 0);
    gemm(P + mp.vW, 128, sd, Nk, 0, P + mp.vb, nullptr, vbuf, Nk, 128, Nk, 128, 0);
    k_mha<<<dim3(Nq, 4), 128, 0, stream>>>(qbuf, kbuf, vbuf, xbuf, Nq, Nk);
    gemm(P + mp.mW, 128, xbuf, Nq, 0, P + mp.mb, nullptr,
         concat + (size_t)128 * Nq, Nq, 128, Nq, 128, 0);
    copyk(qd, concat, 128 * Nq);
    gemm(P + lp.W1, 256, concat, Nq, 0, P + lp.b1, nullptr, hid, Nq, 256, Nq, 256, 1);
    gemm(P + lp.W2, 256, hid, Nq, 0, P + lp.b2, addb, outd, Nq, 128, Nq, 256, 0);
  };
  // cross direction: prob row-softmax(scores), message = value x prob^T, mlp2
  auto cross_dir = [&](const MlpP& lp, const float* scores, const float* qd,
                       const float* value, const float* addb, float* outd) {
    k_row_softmax<<<384, 256, 0, stream>>>(scores, prob, 384);
    gemm(value, 384, prob, 384, 1, nullptr, nullptr,
         concat + (size_t)128 * 384, 384, 128, 384, 384, 0);
    copyk(qd, concat, 49152);
    gemm(P + lp.W1, 256, concat, 384, 0, P + lp.b1, nullptr, hid, 384, 256, 384, 256, 1);
    gemm(P + lp.W2, 256, hid, 384, 0, P + lp.b2, addb, outd, 384, 128, 384, 256, 0);
  };

  hipMemcpyAsync(d0a, in_d0, 49152 * 4, hipMemcpyDeviceToDevice, stream);
  hipMemcpyAsync(d1a, in_d1, 49152 * 4, hipMemcpyDeviceToDevice, stream);
  hipMemcpyAsync(d2a, in_d2, 24576 * 4, hipMemcpyDeviceToDevice, stream);
  hipMemcpyAsync(d3a, in_d3, 24576 * 4, hipMemcpyDeviceToDevice, stream);

  float *c0 = d0a, *c1 = d1a, *c2 = d2a, *c3 = d3a;
  float *n0 = d0b, *n1 = d1b, *n2 = d2b, *n3 = d3b;
  for (int li = 0; li < 4; ++li) {
    if (!is_cross[li]) {
      mha_prop(sa[li], am[li], c0, 384, c0, 384, c0, n0);
      mha_prop(sa[li], am[li], c1, 384, c1, 384, c1, n1);
      mha_prop(pha[li], phm[li], c2, 192, c2, 192, c2, n2);
      mha_prop(gha[li], ghm[li], c3, 192, c3, 192, c3, n3);
    } else {
      const CrossP& cp = ca[li];
      // A/B vectors: W1 = [W1q | W1k | W1d] (feat = [q, k, dist])
      gemm(P + cp.W1, 384, c1, 384, 0, nullptr, nullptr, A1b, 384, 128, 384, 128, 0);
      gemm(P + cp.W1 + 128, 384, c0, 384, 0, nullptr, nullptr, B1b, 384, 128, 384, 128, 0);
      gemm(P + cp.W1, 384, c0, 384, 0, nullptr, nullptr, A0b, 384, 128, 384, 128, 0);
      gemm(P + cp.W1 + 128, 384, c1, 384, 0, nullptr, nullptr, B0b, 384, 128, 384, 128, 0);
      k_cross_score<<<NPAIR / 64, 256, 0, stream>>>(
          P + cp.W1, P + cp.b1, P + cp.W2, P + cp.b2, dist, A1b, B1b, A0b, B0b,
          s1b, s0b);
      cross_dir(am[li], s1b, c1, c0, c1, n1);  // delta1 -> n1
      cross_dir(am[li], s0b, c0, c1, c0, n0);  // delta0 -> n0
      mha_prop(pha[li], phm[li], c2, 192, c1, 384, c2, n2);  // delta21
      mha_prop(pha[li], phm[li], c1, 384, c2, 192, n1, n1);  // += delta12
      mha_prop(gha[li], ghm[li], c0, 384, c3, 192, n0, n0);  // += delta03
      mha_prop(gha[li], ghm[li], c3, 192, c0, 384, c3, n3);  // delta30
    }
    float* t;
    t = c0; c0 = n0; n0 = t;
    t = c1; c1 = n1; n1 = t;
    t = c2; c2 = n2; n2 = t;
    t = c3; c3 = n3; n3 = t;
  }

  // ---- final: log_softmax (axis=-2), Sinkhorn, masking ----
  k_col_logsoftmax<<<384, 256, 0, stream>>>(s1b, sb, 384, 384);
  k_couplings<<<(385 * 385 + 255) / 256, 256, 0, stream>>>(sb, P + bin_off, coup);
  k_zero<<<(770 + 255) / 256, 256, 0, stream>>>(uvec, 770);  // uvec+vvec contiguous
  for (int it = 0; it < 5; ++it) {
    k_sink_row<<<385, 256, 0, stream>>>(coup, vvec, uvec);
    k_sink_col<<<385, 256, 0, stream>>>(coup, uvec, vvec);
  }
  k_zmake<<<NPAIR / 256, 256, 0, stream>>>(coup, uvec, vvec, zb, pmin, pmax);
  k_minmax<<<1, 256, 0, stream>>>(pmin, pmax, 576, sminb);
  k_final<<<NPAIR / 256, 256, 0, stream>>>(zb, sminb, lmb, fidx, unreach, out);
}